// Bert_79817672229408
// MI455X (gfx1250) — compile-verified
//
#include <hip/hip_runtime.h>
#include <math.h>

// ---------------------------------------------------------------------------
// DeBERTa-style disentangled attention layer for MI455X (gfx1250, wave32).
// Matrix math on V_WMMA_F32_16X16X4_F32 (full fp32 precision).
// GEMMs stage tiles HBM -> LDS with gfx1250 async-to-LDS ops (double buffered),
// WMMA fragments come from LDS (ds_load_b64).
// ---------------------------------------------------------------------------

typedef float v2f __attribute__((ext_vector_type(2)));
typedef float v8f __attribute__((ext_vector_type(8)));
typedef int   v4i __attribute__((ext_vector_type(4)));

#define S_LEN   512
#define B_SZ    8
#define NH      12
#define HDIM    64
#define DMODEL  768
#define NBUCKET 63               // 2*BUCKET-1
#define MROWS   (S_LEN * B_SZ)   // 4096 flattened (s,b) rows

static constexpr float SCALE_F = 0.07216878364870322f;   // 1/sqrt(3*64)

#if defined(__has_builtin)
#if __has_builtin(__builtin_amdgcn_global_load_async_to_lds_b128) && \
    __has_builtin(__builtin_amdgcn_s_wait_asynccnt)
#define USE_ASYNC_LDS 1
#endif
#endif

typedef __attribute__((address_space(1))) v4i* gptr_v4i;
typedef __attribute__((address_space(3))) v4i* lptr_v4i;

// Stage 16 bytes global -> LDS (async on gfx1250, sync fallback otherwise).
__device__ __forceinline__ void stage16(const float* g, float* l) {
#ifdef USE_ASYNC_LDS
  __builtin_amdgcn_global_load_async_to_lds_b128(
      (gptr_v4i)(void*)g, (lptr_v4i)(void*)l, 0, 0);
#else
  *(float4*)l = *(const float4*)g;
#endif
}

__device__ __forceinline__ void stage_wait_all() {
#ifdef USE_ASYNC_LDS
  __builtin_amdgcn_s_wait_asynccnt(0);
#endif
}

// D = A(16x4) * B(4x16) + C(16x16), fp32, wave32.
// A layout (ISA 7.12.2): lane l holds row m=l&15, K = 2*(l>>4) + {0,1} in {x,y}.
// C/D layout: lane l -> N = l&15, VGPR r -> M = r + 8*(l>>4).
// B layout mirror of A: lane l holds col n=l&15, K = 2*(l>>4)+{0,1}.
__device__ __forceinline__ v8f wmma_f32_4(v2f a, v2f b, v8f c) {
  return __builtin_amdgcn_wmma_f32_16x16x4_f32(
      /*neg_a=*/false, a, /*neg_b=*/false, b,
      /*c_mod=*/(short)0, c, /*reuse_a=*/false, /*reuse_b=*/false);
}

// _log_bucket from the reference (mid=16, MAXPOS=512, BUCKET=32).
// 4.3305304 = (mid-1) / ln((MAXPOS-1)/mid) = 15 / ln(511/16)
__device__ __forceinline__ int log_bucket(int rp) {
  const int mid = 16;
  int sgn = (rp > 0) - (rp < 0);
  int a = rp < 0 ? -rp : rp;
  int abs_pos = (rp < mid && rp > -mid) ? (mid - 1) : (a < 511 ? a : 511);
  float lg = __logf((float)abs_pos * 0.0625f) * 4.3305304f;
  int log_pos = (int)ceilf(lg) + mid;
  int bucket = (abs_pos <= mid) ? rp : log_pos * sgn;
  return bucket + 31;   // + BUCKET - 1
}

// ---------------------------------------------------------------------------
// Row LayerNorm (no/with affine).  One block per row.
// ---------------------------------------------------------------------------
__global__ __launch_bounds__(256) void row_ln_kernel(
    const float* __restrict__ in, const float* __restrict__ gamma,
    const float* __restrict__ beta, float* __restrict__ out, int ncols) {
  int row = blockIdx.x;
  const float* r = in + (size_t)row * ncols;
  float s = 0.f, s2 = 0.f;
  for (int c = threadIdx.x; c < ncols; c += blockDim.x) {
    float v = r[c]; s += v; s2 += v * v;
  }
  for (int off = 16; off > 0; off >>= 1) {
    s  += __shfl_xor(s,  off, 32);
    s2 += __shfl_xor(s2, off, 32);
  }
  __shared__ float sh[16];
  int lane = threadIdx.x & 31, wid = threadIdx.x >> 5;
  if (lane == 0) { sh[wid] = s; sh[8 + wid] = s2; }
  __syncthreads();
  if (threadIdx.x == 0) {
    float ts = 0.f, ts2 = 0.f;
    for (int i = 0; i < (int)(blockDim.x >> 5); ++i) { ts += sh[i]; ts2 += sh[8 + i]; }
    sh[0] = ts; sh[8] = ts2;
  }
  __syncthreads();
  float mean = sh[0] / (float)ncols;
  float var  = sh[8] / (float)ncols - mean * mean;
  float inv  = rsqrtf(var + 1e-7f);
  for (int c = threadIdx.x; c < ncols; c += blockDim.x) {
    float v = (r[c] - mean) * inv;
    if (gamma) v = v * gamma[c] + beta[c];
    out[(size_t)row * ncols + c] = v;
  }
}

// ---------------------------------------------------------------------------
// Generic batched GEMM:  C[m,n] = alpha * sum_k A[m,k] * Bw[n,k]  (+ bias[n])
// Batch index z = b*NH + h; base pointers offset by (b,h) strides.
// Block = 4 waves = 64(M) x 64(N) tile; K consumed in chunks of 32 staged
// into double-buffered LDS via async global->LDS; fragments via ds_load_b64.
// Edge rows/cols clamped on load, guarded on store.  K % 32 == 0 required.
// ---------------------------------------------------------------------------
#define KC 32
#define LDS_STRIDE 34   // pad: 8B-aligned float2 reads, conflict-free banks

__global__ __launch_bounds__(128) void gemm_wmma_kernel(
    const float* __restrict__ A, long long lda, long long sAb, long long sAh,
    const float* __restrict__ Bw, long long ldb, long long sBb, long long sBh,
    const float* __restrict__ bias,
    float* __restrict__ C, long long ldc, long long sCb, long long sCh,
    int M, int N, int K, float alpha) {
  __shared__ float As[2][64][LDS_STRIDE];
  __shared__ float Bs[2][64][LDS_STRIDE];

  int tid = threadIdx.x;
  int lane = tid & 31, wid = tid >> 5;
  int bb = blockIdx.z / NH, hh = blockIdx.z % NH;
  const float* Ab = A + (long long)bb * sAb + (long long)hh * sAh;
  const float* Bb = Bw + (long long)bb * sBb + (long long)hh * sBh;
  float*       Cb = C + (long long)bb * sCb + (long long)hh * sCh;

  int m0blk = blockIdx.x * 64;
  int n0blk = blockIdx.y * 64;
  int nl = lane & 15;
  int kh = (lane >> 4) << 1;        // K sub-offset for this lane half
  int mhalf = (lane >> 4) << 3;

  // Per-thread staging descriptors: 512 16B-units per matrix, 4 per thread.
  // unit u: row = u>>3, col = (u&7)*4
  int srowA[4], srowB[4], scol[4];
#pragma unroll
  for (int i = 0; i < 4; ++i) {
    int u = tid + i * 128;
    int row = u >> 3;
    scol[i] = (u & 7) << 2;
    int ga = m0blk + row; if (ga > M - 1) ga = M - 1;
    int gb = n0blk + row; if (gb > N - 1) gb = N - 1;
    srowA[i] = ga;
    srowB[i] = gb;
  }

  const v8f vzero = {0.f, 0.f, 0.f, 0.f, 0.f, 0.f, 0.f, 0.f};
  v8f acc[4];
#pragma unroll
  for (int j = 0; j < 4; ++j) acc[j] = vzero;

  int nchunks = K / KC;

  // prologue: stage chunk 0 into buffer 0
#pragma unroll
  for (int i = 0; i < 4; ++i) {
    int u = tid + i * 128;
    int row = u >> 3, col = scol[i];
    stage16(Ab + (long long)srowA[i] * lda + col, &As[0][row][col]);
    stage16(Bb + (long long)srowB[i] * ldb + col, &Bs[0][row][col]);
  }

  for (int kc = 0; kc < nchunks; ++kc) {
    int cur = kc & 1;
    stage_wait_all();
    __syncthreads();
    if (kc + 1 < nchunks) {
      int nxt = 1 - cur;
      int k0 = (kc + 1) * KC;
#pragma unroll
      for (int i = 0; i < 4; ++i) {
        int u = tid + i * 128;
        int row = u >> 3, col = scol[i];
        stage16(Ab + (long long)srowA[i] * lda + k0 + col, &As[nxt][row][col]);
        stage16(Bb + (long long)srowB[i] * ldb + k0 + col, &Bs[nxt][row][col]);
      }
    }
    // consume current chunk from LDS
#pragma unroll
    for (int kt = 0; kt < KC; kt += 4) {
      v2f af = { As[cur][wid * 16 + nl][kt + kh],
                 As[cur][wid * 16 + nl][kt + kh + 1] };
#pragma unroll
      for (int j = 0; j < 4; ++j) {
        v2f bf = { Bs[cur][j * 16 + nl][kt + kh],
                   Bs[cur][j * 16 + nl][kt + kh + 1] };
        acc[j] = wmma_f32_4(af, bf, acc[j]);
      }
    }
  }

#pragma unroll
  for (int j = 0; j < 4; ++j) {
    int n = n0blk + j * 16 + nl;
    if (n >= N) continue;
    float bvv = bias ? bias[n] : 0.f;
#pragma unroll
    for (int r = 0; r < 8; ++r) {
      int m = m0blk + wid * 16 + r + mhalf;
      if (m < M) Cb[(long long)m * ldc + n] = acc[j][r] * alpha + bvv;
    }
  }
}

// ---------------------------------------------------------------------------
// Per-head LayerNorm of q/k halves of qk buffer: [4096, 1536] -> q,k [4096,768]
// ---------------------------------------------------------------------------
__global__ __launch_bounds__(256) void head_ln_kernel(
    const float* __restrict__ qk,
    const float* __restrict__ qg, const float* __restrict__ qb,
    const float* __restrict__ kg, const float* __restrict__ kb,
    float* __restrict__ qout, float* __restrict__ kout) {
  int t = blockIdx.x * blockDim.x + threadIdx.x;
  if (t >= MROWS * NH) return;
  int m = t / NH, h = t % NH;
  const float* qr = qk + (size_t)m * (2 * DMODEL) + h * HDIM;
  const float* kr = qr + DMODEL;

  float s = 0.f, s2 = 0.f;
  for (int d = 0; d < HDIM; ++d) { float v = qr[d]; s += v; s2 += v * v; }
  float mean = s * (1.f / HDIM);
  float inv  = rsqrtf(s2 * (1.f / HDIM) - mean * mean + 1e-7f);
  float* qo = qout + (size_t)m * DMODEL + h * HDIM;
  for (int d = 0; d < HDIM; ++d) qo[d] = (qr[d] - mean) * inv * qg[d] + qb[d];

  s = 0.f; s2 = 0.f;
  for (int d = 0; d < HDIM; ++d) { float v = kr[d]; s += v; s2 += v * v; }
  mean = s * (1.f / HDIM);
  inv  = rsqrtf(s2 * (1.f / HDIM) - mean * mean + 1e-7f);
  float* ko = kout + (size_t)m * DMODEL + h * HDIM;
  for (int d = 0; d < HDIM; ++d) ko[d] = (kr[d] - mean) * inv * kg[d] + kb[d];
}

// ---------------------------------------------------------------------------
// sp[b,h,s] = sigmoid(x[s,b,:].pos_W[h,:] + pos_b[h]) * 1.2 - 0.1
// ---------------------------------------------------------------------------
__global__ __launch_bounds__(256) void pos_sp_kernel(
    const float* __restrict__ x, const float* __restrict__ posW,
    const float* __restrict__ posb, float* __restrict__ sp) {
  int t = blockIdx.x * blockDim.x + threadIdx.x;
  if (t >= MROWS * NH) return;
  int m = t / NH, h = t % NH;
  int s = m >> 3, b = m & 7;
  const float* xr = x + (size_t)m * DMODEL;
  const float* wr = posW + (size_t)h * DMODEL;
  float acc = posb[h];
  for (int d = 0; d < DMODEL; ++d) acc += xr[d] * wr[d];
  float sg = 1.0f / (1.0f + __expf(-acc));
  sp[((size_t)b * NH + h) * S_LEN + s] = sg * 1.2f - 0.1f;
}

__global__ void cum_kernel(const float* __restrict__ sp, float* __restrict__ cum) {
  int z = blockIdx.x * blockDim.x + threadIdx.x;
  if (z >= B_SZ * NH) return;
  const float* r = sp + (size_t)z * S_LEN;
  float* o = cum + (size_t)z * S_LEN;
  float c = 0.f;
  for (int s = 0; s < S_LEN; ++s) { c += r[s]; o[s] = c; }
}

// ---------------------------------------------------------------------------
// Fused attention: scores (WMMA) + dynamic position bias + mask -> LDS,
// wave-cooperative softmax, P.V (WMMA, probs read from LDS) -> ctx.
// 1 wave = 16 query rows; 2 waves/block; LDS 2*16*512*4 = 64 KB.
// grid = (S/32, NH, B)
// ---------------------------------------------------------------------------
__global__ __launch_bounds__(64) void attn_kernel(
    const float* __restrict__ q, const float* __restrict__ k,
    const float* __restrict__ v, const float* __restrict__ qpb,
    const float* __restrict__ kpb, const float* __restrict__ cum,
    const unsigned char* __restrict__ mask, float* __restrict__ ctx) {
  __shared__ float sc[2][16][S_LEN];

  int lane = threadIdx.x & 31, wid = threadIdx.x >> 5;
  int b = blockIdx.z, h = blockIdx.y;
  int q0 = blockIdx.x * 32 + wid * 16;

  const float* cumbh  = cum + ((size_t)b * NH + h) * S_LEN;
  const float* qbase  = q + (size_t)b * DMODEL + h * HDIM;   // row stride B*D
  const float* kbase  = k + (size_t)b * DMODEL + h * HDIM;
  const float* vbase  = v + (size_t)b * DMODEL + h * HDIM;
  const float* qpb_bh = qpb + ((size_t)b * NH + h) * S_LEN * NBUCKET;
  const float* kpb_bh = kpb + ((size_t)b * NH + h) * NBUCKET * S_LEN;
  const unsigned char* mrow = mask + (size_t)b * S_LEN;

  const int ROWSTRIDE = B_SZ * DMODEL;   // 6144
  int nl = lane & 15;
  int kh = (lane >> 4) << 1;
  int mhalf = (lane >> 4) << 3;
  const v8f vzero = {0.f, 0.f, 0.f, 0.f, 0.f, 0.f, 0.f, 0.f};

  const float* arow = qbase + (size_t)(q0 + nl) * ROWSTRIDE;

  // ---- Phase 1: scores tile [16 x 512] ----
  for (int kt = 0; kt < S_LEN / 16; ++kt) {
    int n0 = kt * 16;
    const float* brow = kbase + (size_t)(n0 + nl) * ROWSTRIDE;
    v8f acc = vzero;
#pragma unroll
    for (int kk = 0; kk < HDIM; kk += 4) {
      v2f af = { arow[kk + kh], arow[kk + kh + 1] };
      v2f bf = { brow[kk + kh], brow[kk + kh + 1] };
      acc = wmma_f32_4(af, bf, acc);
    }
    int n = n0 + nl;
    float cumn = cumbh[n];
    bool mk = (mrow[n] != 0);
#pragma unroll
    for (int r = 0; r < 8; ++r) {
      int mloc = r + mhalf;
      int mq = q0 + mloc;
      float sv = acc[r] * SCALE_F;
      // dynamic position bias
      float idxf = cumn - cumbh[mq] + (float)((n > mq) - (n < mq));
      float ff = floorf(idxf);
      int fi = (int)ff;
      int ci = (int)ceilf(idxf);
      float delta = idxf - ff;
      int fb = log_bucket(fi);
      int cb = log_bucket(ci);
      const float* qrow = qpb_bh + (size_t)mq * NBUCKET;
      float fq = qrow[fb], cq = qrow[cb];
      float fk = kpb_bh[fb * S_LEN + n], ck = kpb_bh[cb * S_LEN + n];
      sv += fq + delta * (cq - fq) + fk + delta * (ck - fk);
      if (mk) sv = -__builtin_inff();
      sc[wid][mloc][n] = sv;
    }
  }

  // ---- Phase 2: softmax over K (per-wave LDS region; DS ops in-order) ----
  for (int r = 0; r < 16; ++r) {
    float mx = -__builtin_inff();
    for (int c = lane; c < S_LEN; c += 32) mx = fmaxf(mx, sc[wid][r][c]);
    for (int off = 16; off > 0; off >>= 1) mx = fmaxf(mx, __shfl_xor(mx, off, 32));
    float sum = 0.f;
    for (int c = lane; c < S_LEN; c += 32) {
      float e = __expf(sc[wid][r][c] - mx);
      sc[wid][r][c] = e;
      sum += e;
    }
    for (int off = 16; off > 0; off >>= 1) sum += __shfl_xor(sum, off, 32);
    float inv = 1.0f / sum;
    for (int c = lane; c < S_LEN; c += 32) sc[wid][r][c] *= inv;
  }

  // ---- Phase 3: ctx[16 x 64] = probs[16 x 512] @ v[512 x 64] ----
  for (int nt = 0; nt < 4; ++nt) {
    int n0 = nt * 16;
    v8f acc = vzero;
    for (int kk = 0; kk < S_LEN; kk += 4) {
      v2f af = { sc[wid][nl][kk + kh], sc[wid][nl][kk + kh + 1] };
      v2f bf = { vbase[(size_t)(kk + kh)     * ROWSTRIDE + n0 + nl],
                 vbase[(size_t)(kk + kh + 1) * ROWSTRIDE + n0 + nl] };
      acc = wmma_f32_4(af, bf, acc);
    }
#pragma unroll
    for (int r = 0; r < 8; ++r) {
      int mq = q0 + r + mhalf;
      ctx[((size_t)mq * B_SZ + b) * DMODEL + h * HDIM + n0 + nl] = acc[r];
    }
  }
}

// ---------------------------------------------------------------------------
// Launch
// ---------------------------------------------------------------------------
extern "C" void kernel_launch(void* const* d_in, const int* in_sizes, int n_in,
                              void* d_out, int out_size, void* d_ws, size_t ws_size,
                              hipStream_t stream) {
  (void)in_sizes; (void)n_in; (void)out_size; (void)ws_size;

  const float* hidden = (const float*)d_in[0];
  const float* Wqk    = (const float*)d_in[1];
  const float* bqk    = (const float*)d_in[2];
  const float* Wv     = (const float*)d_in[3];
  const float* bvv    = (const float*)d_in[4];
  const float* Wo     = (const float*)d_in[5];
  const float* bo     = (const float*)d_in[6];
  const float* qg     = (const float*)d_in[7];
  const float* qb     = (const float*)d_in[8];
  const float* kg     = (const float*)d_in[9];
  const float* kb     = (const float*)d_in[10];
  const float* pg     = (const float*)d_in[11];
  const float* pbta   = (const float*)d_in[12];
  const float* posW   = (const float*)d_in[13];
  const float* posb   = (const float*)d_in[14];
  const float* q_rel  = (const float*)d_in[15];
  const float* k_rel  = (const float*)d_in[16];
  const unsigned char* mask = (const unsigned char*)d_in[17];  // jnp bool = 1B

  float* ws    = (float*)d_ws;
  float* x     = ws;                                            // 4096*768
  float* qkbuf = x     + (size_t)MROWS * DMODEL;                // 4096*1536
  float* qln   = qkbuf + (size_t)MROWS * 2 * DMODEL;            // 4096*768
  float* kln   = qln   + (size_t)MROWS * DMODEL;                // 4096*768
  float* vbuf  = kln   + (size_t)MROWS * DMODEL;                // 4096*768
  float* sp    = vbuf  + (size_t)MROWS * DMODEL;                // 96*512
  float* cum   = sp    + (size_t)B_SZ * NH * S_LEN;             // 96*512
  float* qpb   = cum   + (size_t)B_SZ * NH * S_LEN;             // 96*512*63
  float* kpb   = qpb   + (size_t)B_SZ * NH * S_LEN * NBUCKET;   // 96*63*512
  float* ctx   = kpb   + (size_t)B_SZ * NH * NBUCKET * S_LEN;   // 4096*768
  float* proj  = ctx   + (size_t)MROWS * DMODEL;                // 4096*768

  // 1. pre-LN (no affine)
  row_ln_kernel<<<MROWS, 256, 0, stream>>>(hidden, nullptr, nullptr, x, DMODEL);

  // 2. qk = x @ Wqk^T + bqk   [4096, 1536]
  gemm_wmma_kernel<<<dim3(64, 24, 1), 128, 0, stream>>>(
      x, DMODEL, 0, 0, Wqk, DMODEL, 0, 0, bqk,
      qkbuf, 2 * DMODEL, 0, 0, MROWS, 2 * DMODEL, DMODEL, 1.0f);

  // 3. v = x @ Wv^T + bv      [4096, 768]
  gemm_wmma_kernel<<<dim3(64, 12, 1), 128, 0, stream>>>(
      x, DMODEL, 0, 0, Wv, DMODEL, 0, 0, bvv,
      vbuf, DMODEL, 0, 0, MROWS, DMODEL, DMODEL, 1.0f);

  // 4. per-head LN of q/k
  head_ln_kernel<<<(MROWS * NH + 255) / 256, 256, 0, stream>>>(
      qkbuf, qg, qb, kg, kb, qln, kln);

  // 5. dynamic position: sp then cumsum
  pos_sp_kernel<<<(MROWS * NH + 255) / 256, 256, 0, stream>>>(x, posW, posb, sp);
  cum_kernel<<<1, 128, 0, stream>>>(sp, cum);

  // 6. qpb[b,h,q,l] = SCALE * q . k_rel^T   (batched z = b*NH+h; K=64)
  gemm_wmma_kernel<<<dim3(8, 1, B_SZ * NH), 128, 0, stream>>>(
      qln, (long long)B_SZ * DMODEL, DMODEL, HDIM,
      k_rel, (long long)NH * HDIM, 0, HDIM,
      nullptr,
      qpb, NBUCKET, (long long)NH * S_LEN * NBUCKET, (long long)S_LEN * NBUCKET,
      S_LEN, NBUCKET, HDIM, SCALE_F);

  // 7. kpb[b,h,l,k] = SCALE * q_rel . k^T   (batched)
  gemm_wmma_kernel<<<dim3(1, 8, B_SZ * NH), 128, 0, stream>>>(
      q_rel, (long long)NH * HDIM, 0, HDIM,
      kln, (long long)B_SZ * DMODEL, DMODEL, HDIM,
      nullptr,
      kpb, S_LEN, (long long)NH * NBUCKET * S_LEN, (long long)NBUCKET * S_LEN,
      NBUCKET, S_LEN, HDIM, SCALE_F);

  // 8. fused attention (scores + pos bias + mask + softmax + P.V)
  attn_kernel<<<dim3(16, NH, B_SZ), 64, 0, stream>>>(
      qln, kln, vbuf, qpb, kpb, cum, mask, ctx);

  // 9. out projection
  gemm_wmma_kernel<<<dim3(64, 12, 1), 128, 0, stream>>>(
      ctx, DMODEL, 0, 0, Wo, DMODEL, 0, 0, bo,
      proj, DMODEL, 0, 0, MROWS, DMODEL, DMODEL, 1.0f);

  // 10. final LN with affine -> d_out
  row_ln_kernel<<<MROWS, 256, 0, stream>>>(proj, pg, pbta, (float*)d_out, DMODEL);
}